// SubGraph_15350213116562
// MI455X (gfx1250) — compile-verified
//
#include <hip/hip_runtime.h>
#include <math.h>

// ---------------------------------------------------------------------------
// VectorNet SubGraph for MI455X (gfx1250, wave32, WMMA).
// 3x [Linear -> LayerNorm -> ReLU -> Linear -> edge scatter-max -> concat],
// cluster max-pool, column L2 normalize.
// fp32 throughout (memory/atomic-bound workload; V_WMMA_F32_16X16X4_F32 keeps
// reference numerics at zero cost since GEMMs are ~11.5 GFLOP total).
// ---------------------------------------------------------------------------

typedef __attribute__((ext_vector_type(2))) float v2f;
typedef __attribute__((ext_vector_type(8))) float v8f;

#define NEG_INF (-__builtin_inff())

// jnp.where(isfinite(v), v, 0)  -- branch-free: finite*0 == 0, inf/nan*0 == nan
__device__ __forceinline__ float fixf(float v) {
    return (v * 0.0f == 0.0f) ? v : 0.0f;
}

__device__ __forceinline__ void atomic_max_f32(float* p, float v) {
#if defined(__gfx1250__)
    // Native CDNA5 float-max atomic, non-returning (STOREcnt path, no CAS loop)
    asm volatile("global_atomic_max_num_f32 %0, %1, off"
                 :: "v"(p), "v"(v) : "memory");
#else
    atomicMax(p, v);
#endif
}

// ---------------------------------------------------------------------------
// Fused MLP layer: xnext[:, 0:D_OUT]   = (relu(LN(x@w1+b1)))@w2 + b2
//                  xnext[:, D_OUT:2*D_OUT] = -inf   (scatter-max identity)
// 64 node rows per block, 4 waves, each wave owns 16 rows.
// GEMM1 -> LDS -> LayerNorm+ReLU -> GEMM2, all via V_WMMA_F32_16X16X4_F32.
// ---------------------------------------------------------------------------
template<int D_IN, int D_OUT, bool FIX_IN>
__global__ __launch_bounds__(128)
void mlp_layer_kernel(const float* __restrict__ x,
                      const float* __restrict__ w1, const float* __restrict__ b1,
                      const float* __restrict__ g,  const float* __restrict__ be,
                      const float* __restrict__ w2, const float* __restrict__ b2,
                      float* __restrict__ xnext, int N)
{
    constexpr int ROWS = 64;
    constexpr int H    = 64;
    __shared__ float hs[ROWS][H + 4];

    const int t    = threadIdx.x;
    const int wave = t >> 5;
    const int lane = t & 31;
    const int m    = lane & 15;          // col (B/C/D) or row (A) within tile
    const int kh   = (lane >> 4) << 1;   // K sub-offset for f32 A/B frags: 0 or 2
    const int rhal = (lane >> 4) << 3;   // C/D row half: 0 or 8
    const int wr   = wave * 16;          // this wave's row offset in block
    const int block_row = blockIdx.x * ROWS;

    // ---- GEMM1: h = x @ w1, 16x64 per wave, K stepped by 4 ------------------
    v8f acc0 = {}, acc1 = {}, acc2 = {}, acc3 = {};
    {
        const int arow = block_row + wr + m;
        const float* arowp = x + (size_t)(arow < N ? arow : N - 1) * D_IN;
        for (int k0 = 0; k0 < D_IN; k0 += 4) {
            v2f a;
            a.x = arowp[k0 + kh];
            a.y = arowp[k0 + kh + 1];
            if (FIX_IN) { a.x = fixf(a.x); a.y = fixf(a.y); }
            const float* w1k0 = w1 + (size_t)(k0 + kh) * H;
            const float* w1k1 = w1 + (size_t)(k0 + kh + 1) * H;
            v2f b0, b1v, b2v, b3v;
            b0.x  = w1k0[m];      b0.y  = w1k1[m];
            b1v.x = w1k0[16 + m]; b1v.y = w1k1[16 + m];
            b2v.x = w1k0[32 + m]; b2v.y = w1k1[32 + m];
            b3v.x = w1k0[48 + m]; b3v.y = w1k1[48 + m];
            acc0 = __builtin_amdgcn_wmma_f32_16x16x4_f32(false, a, false, b0,  (short)0, acc0, false, false);
            acc1 = __builtin_amdgcn_wmma_f32_16x16x4_f32(false, a, false, b1v, (short)0, acc1, false, false);
            acc2 = __builtin_amdgcn_wmma_f32_16x16x4_f32(false, a, false, b2v, (short)0, acc2, false, false);
            acc3 = __builtin_amdgcn_wmma_f32_16x16x4_f32(false, a, false, b3v, (short)0, acc3, false, false);
        }
    }
    // spill h tile (+bias) to LDS so LayerNorm can see whole rows
    #pragma unroll
    for (int r = 0; r < 8; ++r) {
        hs[wr + rhal + r][ 0 + m] = acc0[r] + b1[ 0 + m];
        hs[wr + rhal + r][16 + m] = acc1[r] + b1[16 + m];
        hs[wr + rhal + r][32 + m] = acc2[r] + b1[32 + m];
        hs[wr + rhal + r][48 + m] = acc3[r] + b1[48 + m];
    }
    __syncthreads();

    // ---- LayerNorm + ReLU: one thread per row ------------------------------
    if (t < ROWS) {
        float mu = 0.f;
        #pragma unroll
        for (int c = 0; c < H; ++c) mu += hs[t][c];
        mu *= (1.0f / H);
        float var = 0.f;
        #pragma unroll
        for (int c = 0; c < H; ++c) { float d = hs[t][c] - mu; var += d * d; }
        var *= (1.0f / H);
        const float rs = rsqrtf(var + 1e-5f);
        #pragma unroll
        for (int c = 0; c < H; ++c) {
            float v = (hs[t][c] - mu) * rs * g[c] + be[c];
            hs[t][c] = v > 0.f ? v : 0.f;
        }
    }
    __syncthreads();

    // ---- GEMM2: xt = h @ w2 + b2 -> left half; -inf -> right half ----------
    const int grow_base = block_row + wr;
    for (int ct = 0; ct < D_OUT / 16; ++ct) {
        v8f acc = {};
        for (int k0 = 0; k0 < H; k0 += 4) {
            v2f a;
            a.x = hs[wr + m][k0 + kh];
            a.y = hs[wr + m][k0 + kh + 1];
            v2f b;
            b.x = w2[(size_t)(k0 + kh) * D_OUT + ct * 16 + m];
            b.y = w2[(size_t)(k0 + kh + 1) * D_OUT + ct * 16 + m];
            acc = __builtin_amdgcn_wmma_f32_16x16x4_f32(false, a, false, b, (short)0, acc, false, false);
        }
        const float bias = b2[ct * 16 + m];
        #pragma unroll
        for (int r = 0; r < 8; ++r) {
            const int grow = grow_base + rhal + r;
            if (grow < N) {
                float* rowp = xnext + (size_t)grow * (2 * D_OUT);
                rowp[ct * 16 + m]         = acc[r] + bias;
                rowp[D_OUT + ct * 16 + m] = NEG_INF;
            }
        }
    }
}

// ---------------------------------------------------------------------------
// Edge scatter-max: xn[dst, D + c] = max over edges of xn[src, c]
// One thread per (edge, 4 channels): float4 gather + 4 native float atomics.
// ---------------------------------------------------------------------------
template<int D>
__global__ __launch_bounds__(256)
void edge_scatter_max_kernel(const int* __restrict__ ei, float* __restrict__ xn,
                             int E)
{
    constexpr int G = D / 4;
    const long long tid = (long long)blockIdx.x * blockDim.x + threadIdx.x;
    if (tid >= (long long)E * G) return;
    const int e = (int)(tid / G);
    const int c = (int)(tid % G) * 4;
    const int src = ei[e];
    const int dst = ei[E + e];
    const float4 v = *(const float4*)(xn + (size_t)src * (2 * D) + c);
    float* dp = xn + (size_t)dst * (2 * D) + D + c;
    atomic_max_f32(dp + 0, v.x);
    atomic_max_f32(dp + 1, v.y);
    atomic_max_f32(dp + 2, v.z);
    atomic_max_f32(dp + 3, v.w);
}

// ---------------------------------------------------------------------------
// Fill buffer with -inf (pool identity)
// ---------------------------------------------------------------------------
__global__ __launch_bounds__(256)
void fill_neginf_kernel(float* __restrict__ p, int total)
{
    const int i = blockIdx.x * blockDim.x + threadIdx.x;
    if (i < total) p[i] = NEG_INF;
}

// ---------------------------------------------------------------------------
// Cluster max-pool: pooled[cluster[n], c] = max over n of fix(x3[n, c])
// ---------------------------------------------------------------------------
__global__ __launch_bounds__(256)
void pool_kernel(const float* __restrict__ x3, const int* __restrict__ cluster,
                 float* __restrict__ pooled, int N)
{
    constexpr int FD = 512, G = FD / 4;
    const long long tid = (long long)blockIdx.x * blockDim.x + threadIdx.x;
    if (tid >= (long long)N * G) return;
    const int n = (int)(tid / G);
    const int c = (int)(tid % G) * 4;
    float4 v = *(const float4*)(x3 + (size_t)n * FD + c);
    v.x = fixf(v.x); v.y = fixf(v.y); v.z = fixf(v.z); v.w = fixf(v.w);
    float* dp = pooled + (size_t)cluster[n] * FD + c;
    atomic_max_f32(dp + 0, v.x);
    atomic_max_f32(dp + 1, v.y);
    atomic_max_f32(dp + 2, v.z);
    atomic_max_f32(dp + 3, v.w);
}

// ---------------------------------------------------------------------------
// Per-column: fix non-finite -> 0 in place, norms[c] = ||col||_2 (0 -> 1)
// ---------------------------------------------------------------------------
__global__ __launch_bounds__(256)
void col_norm_kernel(float* __restrict__ pooled, float* __restrict__ norms, int P)
{
    constexpr int FD = 512;
    const int c = blockIdx.x;
    const int t = threadIdx.x;
    float s = 0.f;
    for (int r = t; r < P; r += 256) {
        float v = fixf(pooled[(size_t)r * FD + c]);
        pooled[(size_t)r * FD + c] = v;
        s += v * v;
    }
    __shared__ float red[256];
    red[t] = s;
    __syncthreads();
    for (int o = 128; o > 0; o >>= 1) {
        if (t < o) red[t] += red[t + o];
        __syncthreads();
    }
    if (t == 0) {
        const float nv = sqrtf(red[0]);
        norms[c] = (nv == 0.f) ? 1.f : nv;
    }
}

__global__ __launch_bounds__(256)
void scale_kernel(float* __restrict__ pooled, const float* __restrict__ norms,
                  int total)
{
    const int i = blockIdx.x * blockDim.x + threadIdx.x;
    if (i < total) pooled[i] /= norms[i & 511];
}

// ---------------------------------------------------------------------------

extern "C" void kernel_launch(void* const* d_in, const int* in_sizes, int n_in,
                              void* d_out, int out_size, void* d_ws, size_t ws_size,
                              hipStream_t stream)
{
    const float* x0      = (const float*)d_in[0];
    const int*   ei      = (const int*)  d_in[1];
    const int*   cluster = (const int*)  d_in[2];
    const int N = in_sizes[0] / 64;
    const int E = in_sizes[1] / 2;
    const int P = out_size / 512;

    // per-layer params: w1, b1, g, be, w2, b2 starting at index 3
    const float* prm[3][6];
    for (int i = 0; i < 3; ++i)
        for (int j = 0; j < 6; ++j)
            prm[i][j] = (const float*)d_in[3 + 6 * i + j];

    float* ws    = (float*)d_ws;
    float* x1    = ws;                       // N * 128
    float* x2    = x1 + (size_t)N * 128;     // N * 256
    float* x3    = x2 + (size_t)N * 256;     // N * 512
    float* norms = x3 + (size_t)N * 512;     // 512
    float* pooled = (float*)d_out;           // P * 512 (output, built in place)

    const int nblocks = (N + 63) / 64;
    const dim3 blk(128);

    // layer 0: 64 -> concat 128
    mlp_layer_kernel<64, 64, false><<<nblocks, blk, 0, stream>>>(
        x0, prm[0][0], prm[0][1], prm[0][2], prm[0][3], prm[0][4], prm[0][5], x1, N);
    {
        const long long total = (long long)E * (64 / 4);
        edge_scatter_max_kernel<64><<<(int)((total + 255) / 256), 256, 0, stream>>>(ei, x1, E);
    }
    // layer 1: 128 -> concat 256 (fix -inf empties of layer0 on load)
    mlp_layer_kernel<128, 128, true><<<nblocks, blk, 0, stream>>>(
        x1, prm[1][0], prm[1][1], prm[1][2], prm[1][3], prm[1][4], prm[1][5], x2, N);
    {
        const long long total = (long long)E * (128 / 4);
        edge_scatter_max_kernel<128><<<(int)((total + 255) / 256), 256, 0, stream>>>(ei, x2, E);
    }
    // layer 2: 256 -> concat 512
    mlp_layer_kernel<256, 256, true><<<nblocks, blk, 0, stream>>>(
        x2, prm[2][0], prm[2][1], prm[2][2], prm[2][3], prm[2][4], prm[2][5], x3, N);
    {
        const long long total = (long long)E * (256 / 4);
        edge_scatter_max_kernel<256><<<(int)((total + 255) / 256), 256, 0, stream>>>(ei, x3, E);
    }

    // cluster max-pool into d_out, then column L2 normalize
    const int ptotal = P * 512;
    fill_neginf_kernel<<<(ptotal + 255) / 256, 256, 0, stream>>>(pooled, ptotal);
    {
        const long long total = (long long)N * (512 / 4);
        pool_kernel<<<(int)((total + 255) / 256), 256, 0, stream>>>(x3, cluster, pooled, N);
    }
    col_norm_kernel<<<512, 256, 0, stream>>>(pooled, norms, P);
    scale_kernel<<<(ptotal + 255) / 256, 256, 0, stream>>>(pooled, norms, ptotal);
}